// GAT_83382495084588
// MI455X (gfx1250) — compile-verified
//
#include <hip/hip_runtime.h>

#define NN 50000
#define EE 850000
#define NEG_SLOPE 0.2f

typedef __attribute__((ext_vector_type(2))) float v2f;
typedef __attribute__((ext_vector_type(8))) float v8f;

// ---- monotonic float <-> ordered-uint encoding (for atomic segment max) ----
__device__ __forceinline__ unsigned f2ord(float f) {
  unsigned u = __float_as_uint(f);
  return (u & 0x80000000u) ? ~u : (u | 0x80000000u);
}
__device__ __forceinline__ float ord2f(unsigned u) {
  return (u & 0x80000000u) ? __uint_as_float(u & 0x7fffffffu)
                           : __uint_as_float(~u);
}

// ---- CDNA5 async global->LDS copy (ASYNCcnt-tracked, bypasses VGPRs) ------
__device__ __forceinline__ void async_copy_b128(unsigned lds_off, const float* gsrc) {
  asm volatile("global_load_async_to_lds_b128 %0, %1, off"
               :: "v"(lds_off), "v"(gsrc) : "memory");
}
__device__ __forceinline__ void wait_asynccnt0() {
  asm volatile("s_wait_asynccnt 0" ::: "memory");
}

// ---------------------------------------------------------------------------
// C[16-row stripe] = A(16xK, staged in LDS via async copy) * B(KxNcols),
// fp32 WMMA 16x16x4.  blockDim.x = 2*Ncols (one wave per 16-col tile).
// gridDim.x = N/16.  LDS row stride = K+4 dwords (16B-aligned rows, and
// 260 mod 64 = 4 -> 16 rows hit 16 distinct banks: conflict-free).
// A fragment (16x4 f32): lanes 0-15 = M, VGPR pair = {K0,K1} (lane-half adds 2).
// B fragment (4x16 f32): lanes hold N, same K split. C/D: VGPR v -> row v + 8*kh.
// ---------------------------------------------------------------------------
__global__ void __launch_bounds__(512)
gemm16_wmma_f32(const float* __restrict__ A, const float* __restrict__ B,
                float* __restrict__ C, int K, int Ncols) {
  extern __shared__ float Atile[];            // 16 x (K+4)
  const int ldA = K + 4;
  const int m0  = blockIdx.x * 16;

  // stage A tile: 16B chunks, async global->LDS
  const int cpr = K >> 2;                     // 4-float chunks per row
  for (int idx = threadIdx.x; idx < 16 * cpr; idx += blockDim.x) {
    int r = idx / cpr, c = (idx - r * cpr) << 2;
    unsigned lds_off = (unsigned)(size_t)&Atile[r * ldA + c];
    async_copy_b128(lds_off, A + (size_t)(m0 + r) * K + c);
  }
  wait_asynccnt0();
  __syncthreads();

  const int wave = threadIdx.x >> 5;
  const int lane = threadIdx.x & 31;
  const int m    = lane & 15;                 // row (A) / column-in-tile (B,C)
  const int kh   = lane >> 4;                 // lane-half -> K offset of 2
  const int ncol = wave * 16 + m;

  v8f acc = {};
  for (int kb = 0; kb < K; kb += 4) {
    const int k0 = kb + 2 * kh;
    v2f a, b;
    a.x = Atile[m * ldA + k0];
    a.y = Atile[m * ldA + k0 + 1];
    b.x = B[(size_t)k0 * Ncols + ncol];
    b.y = B[(size_t)(k0 + 1) * Ncols + ncol];
    acc = __builtin_amdgcn_wmma_f32_16x16x4_f32(false, a, false, b,
                                                (short)0, acc, false, false);
  }
#pragma unroll
  for (int v = 0; v < 8; ++v) {
    const int row = m0 + v + 8 * kh;
    C[(size_t)row * Ncols + ncol] = acc[v];
  }
}

// el[n,h] = <feat[n,h,:], al[h,:]> ; er likewise.
__global__ void dot_el_er(const float* __restrict__ feat, const float* __restrict__ al,
                          const float* __restrict__ ar, float* __restrict__ el,
                          float* __restrict__ er, int H, int D) {
  int t = blockIdx.x * blockDim.x + threadIdx.x;
  if (t >= NN * H) return;
  int n = t / H, h = t - n * H;
  const float* f = feat + (size_t)(n * H + h) * D;
  const float* a = al + h * D;
  const float* b = ar + h * D;
  float sl = 0.f, sr = 0.f;
  for (int d = 0; d < D; ++d) { float v = f[d]; sl += v * a[d]; sr += v * b[d]; }
  el[t] = sl;
  er[t] = sr;
}

// Pass 1: e = leaky_relu(el[src]+er[dst]); atomic segment max over dst.
__global__ void edge_pass1(const float* __restrict__ el, const float* __restrict__ er,
                           const int* __restrict__ src, const int* __restrict__ dst,
                           float* __restrict__ ebuf, unsigned* __restrict__ mbuf, int H) {
  int t = blockIdx.x * blockDim.x + threadIdx.x;
  if (t >= EE * H) return;
  int i = t / H, h = t - i * H;
  float e = el[src[i] * H + h] + er[dst[i] * H + h];
  e = e > 0.f ? e : NEG_SLOPE * e;
  ebuf[t] = e;
  atomicMax(&mbuf[dst[i] * H + h], f2ord(e));
}

// Pass 2: ex = exp(e - max); atomic segment sum of ex (denominator).
__global__ void edge_pass2(const int* __restrict__ dst, float* __restrict__ ebuf,
                           const unsigned* __restrict__ mbuf, float* __restrict__ den,
                           int H) {
  int t = blockIdx.x * blockDim.x + threadIdx.x;
  if (t >= EE * H) return;
  int i = t / H, h = t - i * H;
  int dj = dst[i] * H + h;
  float ex = __expf(ebuf[t] - ord2f(mbuf[dj]));
  ebuf[t] = ex;
  atomicAdd(&den[dj], ex);
}

// Pass 3: agg[dst,h,:] += feat[src,h,:] * (ex / den[dst,h]).
__global__ void edge_agg(const float* __restrict__ ebuf, const float* __restrict__ den,
                         const float* __restrict__ feat, float* __restrict__ agg,
                         const int* __restrict__ src, const int* __restrict__ dst,
                         int H, int D) {
  int t = blockIdx.x * blockDim.x + threadIdx.x;
  if (t >= EE * D) return;
  int i = t / D, d = t - i * D;
  int s = src[i], dn = dst[i];
  for (int h = 0; h < H; ++h) {
    float alpha = ebuf[i * H + h] / den[dn * H + h];
    atomicAdd(&agg[(size_t)(dn * H + h) * D + d],
              feat[(size_t)(s * H + h) * D + d] * alpha);
  }
}

// h1 = elu(agg1 + b1)  (in place, C = 256)
__global__ void bias_elu(float* __restrict__ h1, const float* __restrict__ b1) {
  int t = blockIdx.x * blockDim.x + threadIdx.x;
  if (t >= NN * 256) return;
  float x = h1[t] + b1[t & 255];
  h1[t] = x > 0.f ? x : (__expf(x) - 1.f);
}

// out = agg2 + b2  (C = 64; mean over the single head is identity)
__global__ void final_bias(const float* __restrict__ agg2, const float* __restrict__ b2,
                           float* __restrict__ out) {
  int t = blockIdx.x * blockDim.x + threadIdx.x;
  if (t >= NN * 64) return;
  out[t] = agg2[t] + b2[t & 63];
}

extern "C" void kernel_launch(void* const* d_in, const int* in_sizes, int n_in,
                              void* d_out, int out_size, void* d_ws, size_t ws_size,
                              hipStream_t stream) {
  (void)in_sizes; (void)n_in; (void)out_size; (void)ws_size;
  const float* h   = (const float*)d_in[0];
  const float* W1  = (const float*)d_in[1];
  const float* al1 = (const float*)d_in[2];
  const float* ar1 = (const float*)d_in[3];
  const float* b1  = (const float*)d_in[4];
  const float* W2  = (const float*)d_in[5];
  const float* al2 = (const float*)d_in[6];
  const float* ar2 = (const float*)d_in[7];
  const float* b2  = (const float*)d_in[8];
  const int*   src = (const int*)d_in[9];
  const int*   dst = (const int*)d_in[10];
  float* out = (float*)d_out;

  float* ws = (float*)d_ws;
  float*    feat1 = ws;            ws += (size_t)NN * 256;
  float*    h1    = ws;            ws += (size_t)NN * 256;   // agg1 then elu'd in place
  float*    e1    = ws;            ws += (size_t)EE * 8;
  unsigned* m1    = (unsigned*)ws; ws += (size_t)NN * 8;
  float*    den1  = ws;            ws += (size_t)NN * 8;
  float*    el1   = ws;            ws += (size_t)NN * 8;
  float*    er1   = ws;            ws += (size_t)NN * 8;
  float*    feat2 = ws;            ws += (size_t)NN * 64;
  float*    e2    = ws;            ws += (size_t)EE;
  unsigned* m2    = (unsigned*)ws; ws += (size_t)NN;
  float*    den2  = ws;            ws += (size_t)NN;
  float*    el2   = ws;            ws += (size_t)NN;
  float*    er2   = ws;            ws += (size_t)NN;
  float*    agg2  = ws;            ws += (size_t)NN * 64;

  // zero accumulators each call (graph-capture-safe stream memsets)
  hipMemsetAsync(h1,   0, (size_t)NN * 256 * 4, stream);
  hipMemsetAsync(m1,   0, (size_t)NN * 8 * 2 * 4, stream);  // m1 + den1 (contiguous)
  hipMemsetAsync(m2,   0, (size_t)NN * 2 * 4, stream);      // m2 + den2 (contiguous)
  hipMemsetAsync(agg2, 0, (size_t)NN * 64 * 4, stream);

  const int shmem = 16 * (256 + 4) * 4;

  // ---------------- layer 1 (H=8, D=32) ----------------
  gemm16_wmma_f32<<<NN / 16, 512, shmem, stream>>>(h, W1, feat1, 256, 256);
  dot_el_er<<<(NN * 8 + 255) / 256, 256, 0, stream>>>(feat1, al1, ar1, el1, er1, 8, 32);
  edge_pass1<<<(EE * 8 + 255) / 256, 256, 0, stream>>>(el1, er1, src, dst, e1, m1, 8);
  edge_pass2<<<(EE * 8 + 255) / 256, 256, 0, stream>>>(dst, e1, m1, den1, 8);
  edge_agg<<<(EE * 32 + 255) / 256, 256, 0, stream>>>(e1, den1, feat1, h1, src, dst, 8, 32);
  bias_elu<<<(NN * 256 + 255) / 256, 256, 0, stream>>>(h1, b1);

  // ---------------- layer 2 (H=1, D=64) ----------------
  gemm16_wmma_f32<<<NN / 16, 128, shmem, stream>>>(h1, W2, feat2, 256, 64);
  dot_el_er<<<(NN + 255) / 256, 256, 0, stream>>>(feat2, al2, ar2, el2, er2, 1, 64);
  edge_pass1<<<(EE + 255) / 256, 256, 0, stream>>>(el2, er2, src, dst, e2, m2, 1);
  edge_pass2<<<(EE + 255) / 256, 256, 0, stream>>>(dst, e2, m2, den2, 1);
  edge_agg<<<(EE * 64 + 255) / 256, 256, 0, stream>>>(e2, den2, feat2, agg2, src, dst, 1, 64);
  final_bias<<<(NN * 64 + 255) / 256, 256, 0, stream>>>(agg2, b2, out);
}